// PointNetFeaturePropagation_42880953483444
// MI455X (gfx1250) — compile-verified
//
#include <hip/hip_runtime.h>
#include <hip/hip_bf16.h>

// ---------------------------------------------------------------------------
// PointNet Feature Propagation on gfx1250 (MI455X), fp32 WMMA path.
//   B=16, N=4096, S=1024, D=128, M0=256, M1=128, C_IN=2N=8192
//   out[b] = W2 @ (W1 @ [f1[b]; interp[b]] + b1) + b2   (per-batch GEMMs)
//
// Pipeline:
//   1) nn_idx_kernel        : 3-NN indices per query point  -> ws.idx
//   2) gemm_partial<256,..> : split-K (x16) W1 GEMM, interp rows gathered
//                             on the fly from features2     -> ws.P (partials)
//   3) reduce_bias_kernel   : fixed-order partial sum + b1  -> ws.H
//   4) gemm_partial<128,..> : W2 GEMM + b2                  -> d_out
// Deterministic: no atomics; partials summed in fixed order.
// ---------------------------------------------------------------------------

typedef float v2f __attribute__((ext_vector_type(2)));
typedef float v8f __attribute__((ext_vector_type(8)));

static constexpr int kB = 16;
static constexpr int kN = 4096;
static constexpr int kS = 1024;
static constexpr int kD = 128;
static constexpr int kM0 = 256;
static constexpr int kKSPLIT = 16;  // K-splits for GEMM1 -> 256 workgroups

// ---------------------------------------------------------------------------
// Kernel 1: 3-NN search. Block = 256 threads = 256 query points of one batch.
// xyz2[b] (+ squared norms) cached in LDS (16 KB).
// ---------------------------------------------------------------------------
__global__ __launch_bounds__(256)
void nn_idx_kernel(const float* __restrict__ xyz1,
                   const float* __restrict__ xyz2,
                   int* __restrict__ idx_out) {
  __shared__ float4 p2s[kS];  // x, y, z, ||p||^2

  const int b   = blockIdx.y;
  const int tid = threadIdx.x;

  const float* x2b = xyz2 + (size_t)b * kS * 3;
  for (int i = tid; i < kS; i += 256) {
    float x = x2b[i * 3 + 0];
    float y = x2b[i * 3 + 1];
    float z = x2b[i * 3 + 2];
    p2s[i] = make_float4(x, y, z, x * x + y * y + z * z);
  }
  __syncthreads();

  // same ||a||^2+||b||^2-2ab formula as reference; sqrt is monotone so the
  // pre-sqrt compare yields the same top-3; strict '<' keeps earlier index
  // on ties, matching top_k.
  const int n = blockIdx.x * 256 + tid;
  const float* p1 = xyz1 + ((size_t)b * kN + n) * 3;
  const float ax = p1[0], ay = p1[1], az = p1[2];
  const float sq1 = ax * ax + ay * ay + az * az;

  float d0 = 3.4e38f, d1 = 3.4e38f, d2 = 3.4e38f;
  int   i0 = 0, i1 = 0, i2 = 0;
  for (int s = 0; s < kS; ++s) {
    float4 q = p2s[s];
    float d = sq1 + q.w - 2.0f * (ax * q.x + ay * q.y + az * q.z);
    if (d < d0)      { d2 = d1; i2 = i1; d1 = d0; i1 = i0; d0 = d; i0 = s; }
    else if (d < d1) { d2 = d1; i2 = i1; d1 = d;  i1 = s; }
    else if (d < d2) { d2 = d;  i2 = s; }
  }
  int* o = idx_out + ((size_t)b * kN + n) * 3;
  o[0] = i0; o[1] = i1; o[2] = i2;
}

// ---------------------------------------------------------------------------
// Split-K GEMM partial:  P[s][b] = W[:, ks:ke] @ X[b][ks:ke, :]  (M x 128)
// X rows [0, K0) come from X0[b]; rows [K0, K) are gathered on the fly as the
// mean of 3 features2 rows (GATHER=true) -- fuses the interp step, features2
// is L2-resident so the 3x re-read never touches HBM.
//
// Block = (M/16) waves; wave w owns output rows [16w,16w+16) x 128 cols
// (8 fp32 accumulator tiles).  K staged through LDS in KC=32 chunks:
//   Ws[M][33]  : stride 33 -> conflict-free 16-row column reads (A frags)
//   Xs[32][132]: stride 132 -> 16B-aligned float4 stores, clean row reads (B)
// A fragment per ISA layout of V_WMMA_F32_16X16X4_F32:
//   m = lane&15 is the matrix row, half = lane>>4 selects K {0,1} vs {2,3}.
// ---------------------------------------------------------------------------
template <int M, int K, int KSPLIT, bool GATHER>
__global__ __launch_bounds__(M * 2)
void gemm_partial_kernel(const float* __restrict__ W,
                         const float* __restrict__ X0, long x0_bstride, int K0,
                         const float* __restrict__ G,      // features2 (gather)
                         const int* __restrict__ gidx,     // 3-NN ids (gather)
                         const float* __restrict__ bias,   // nullptr -> none
                         float* __restrict__ P) {
  constexpr int KC    = 32;
  constexpr int NCOL  = 128;
  constexpr int NT    = M * 2;
  constexpr int KSPAN = K / KSPLIT;

  __shared__ __align__(16) float Ws[M][KC + 1];
  __shared__ __align__(16) float Xs[KC][132];

  const int b   = blockIdx.x;
  const int ks  = blockIdx.y * KSPAN;
  const int tid = threadIdx.x;
  const float* X0b = X0 + (long)b * x0_bstride;
  const float* Gb  = GATHER ? (G + (long)b * kS * kD) : nullptr;

  const int lane = tid & 31;
  const int wave = tid >> 5;
  const int r0   = wave * 16;
  const int m    = lane & 15;
  const int half = lane >> 4;
  const int kk2  = half * 2;

  v8f acc[8] = {};

  for (int k = ks; k < ks + KSPAN; k += KC) {
    // ---- stage W[:, k:k+KC] (coalesced float4 loads) ----------------------
#pragma unroll
    for (int i = 0; i < (M * KC / 4) / NT; ++i) {
      int f   = i * NT + tid;
      int row = f >> 3;  // 8 float4 per row of KC=32
      int c4  = f & 7;
      const float* src = W + (long)row * K + k + c4 * 4;
      float4 v = *(const float4*)src;
      if (k + KC < K) __builtin_prefetch(src + KC, 0, 1);  // global_prefetch_b8
      float* dst = &Ws[row][c4 * 4];
      dst[0] = v.x; dst[1] = v.y; dst[2] = v.z; dst[3] = v.w;
    }
    // ---- stage X[k:k+KC, 0:128]; gather-average for concat rows -----------
#pragma unroll
    for (int i = 0; i < (KC * NCOL / 4) / NT; ++i) {
      int f   = i * NT + tid;
      int row = f >> 5;  // 32 float4 per row of 128
      int c4  = f & 31;
      int kr  = k + row;
      float4 v;
      if (!GATHER || kr < K0) {
        v = *(const float4*)(X0b + (long)kr * NCOL + c4 * 4);
      } else {
        int p = kr - K0;
        const int* ip = gidx + ((long)b * kN + p) * 3;
        const float* r0p = Gb + (long)ip[0] * kD + c4 * 4;
        const float* r1p = Gb + (long)ip[1] * kD + c4 * 4;
        const float* r2p = Gb + (long)ip[2] * kD + c4 * 4;
        float4 a0 = *(const float4*)r0p;
        float4 a1 = *(const float4*)r1p;
        float4 a2 = *(const float4*)r2p;
        v = make_float4((a0.x + a1.x + a2.x) * (1.0f / 3.0f),
                        (a0.y + a1.y + a2.y) * (1.0f / 3.0f),
                        (a0.z + a1.z + a2.z) * (1.0f / 3.0f),
                        (a0.w + a1.w + a2.w) * (1.0f / 3.0f));
      }
      *(float4*)&Xs[row][c4 * 4] = v;
    }
    __syncthreads();

    // ---- 8 k-steps x 8 n-tiles of V_WMMA_F32_16X16X4_F32 ------------------
#pragma unroll
    for (int kk = 0; kk < KC; kk += 4) {
      v2f a;
      a.x = Ws[r0 + m][kk + kk2];
      a.y = Ws[r0 + m][kk + kk2 + 1];
#pragma unroll
      for (int t = 0; t < 8; ++t) {
        v2f bb;
        bb.x = Xs[kk + kk2][t * 16 + m];
        bb.y = Xs[kk + kk2 + 1][t * 16 + m];
        acc[t] = __builtin_amdgcn_wmma_f32_16x16x4_f32(
            false, a, false, bb, (short)0, acc[t], false, false);
      }
    }
    __syncthreads();
  }

  // ---- epilogue: C/D layout -> row = r0 + j + 8*half, col = 16t + m -------
  float* Pb = P + ((long)blockIdx.y * gridDim.x + b) * M * NCOL;
#pragma unroll
  for (int t = 0; t < 8; ++t) {
#pragma unroll
    for (int j = 0; j < 8; ++j) {
      int row = r0 + j + half * 8;
      int col = t * 16 + m;
      float v = acc[t][j];
      if (bias) v += bias[row];
      Pb[row * NCOL + col] = v;
    }
  }
}

// ---------------------------------------------------------------------------
// Kernel 3: H = sum over 16 K-splits (fixed order => deterministic) + b1.
// ---------------------------------------------------------------------------
__global__ __launch_bounds__(256)
void reduce_bias_kernel(const float* __restrict__ P,
                        const float* __restrict__ bias,
                        float* __restrict__ H) {
  const long i = (long)blockIdx.x * 256 + threadIdx.x;  // over B*256*128
  constexpr long stride = (long)kB * kM0 * kD;
  float s = 0.0f;
#pragma unroll
  for (int j = 0; j < kKSPLIT; ++j) s += P[(long)j * stride + i];
  int r = (int)(i >> 7) & (kM0 - 1);
  H[i] = s + bias[r];
}

// ---------------------------------------------------------------------------
extern "C" void kernel_launch(void* const* d_in, const int* in_sizes, int n_in,
                              void* d_out, int out_size, void* d_ws,
                              size_t ws_size, hipStream_t stream) {
  const float* xyz1 = (const float*)d_in[0];
  const float* xyz2 = (const float*)d_in[1];
  const float* f1   = (const float*)d_in[2];
  const float* f2   = (const float*)d_in[3];
  const float* w1   = (const float*)d_in[4];
  const float* b1   = (const float*)d_in[5];
  const float* w2   = (const float*)d_in[6];
  const float* b2   = (const float*)d_in[7];

  // workspace layout: idx [B,N,3] ints (0.75 MB) | P partials (32 MB) | H (2 MB)
  char* ws = (char*)d_ws;
  int*   idxbuf = (int*)ws;
  size_t off    = (size_t)kB * kN * 3 * sizeof(int);        // 786432 (256-aligned)
  float* P      = (float*)(ws + off);
  off += (size_t)kKSPLIT * kB * kM0 * kD * sizeof(float);   // 32 MB
  float* H      = (float*)(ws + off);                       // 2 MB

  // 1) 3-NN indices
  nn_idx_kernel<<<dim3(kN / 256, kB), 256, 0, stream>>>(xyz1, xyz2, idxbuf);

  // 2) split-K W1 GEMM with fused interp gather: 16 batches x 16 K-splits
  gemm_partial_kernel<kM0, 2 * kN, kKSPLIT, true>
      <<<dim3(kB, kKSPLIT), kM0 * 2, 0, stream>>>(
          w1, f1, (long)kN * kD, kN, f2, idxbuf, nullptr, P);

  // 3) partial reduction + b1 -> H [B,256,128]
  reduce_bias_kernel<<<(kB * kM0 * kD) / 256, 256, 0, stream>>>(P, b1, H);

  // 4) out[b] = W2 @ H[b] + b2  (128 x 256 x 128 per batch)
  gemm_partial_kernel<128, 256, 1, false>
      <<<dim3(kB, 1), 256, 0, stream>>>(
          w2, H, (long)kM0 * kD, 256, nullptr, nullptr, b2, (float*)d_out);
}